// GridDecDeepVPN_33328946217238
// MI455X (gfx1250) — compile-verified
//
#include <hip/hip_runtime.h>

typedef __attribute__((ext_vector_type(2))) float v2f;
typedef __attribute__((ext_vector_type(8))) float v8f;

#define SIDE   9
#define STATE  81
#define NA     5
#define KPAD   84      // 81 padded up to multiple of 4
#define TILE_B 96      // rows per block (6 waves * 16)
#define THREADS 192

__global__ __launch_bounds__(THREADS) void vpn_kernel(
    const float* __restrict__ obs,   // [B, 243]
    const float* __restrict__ act,   // [B, 81, 5]
    const float* __restrict__ W1,    // [81, 32]
    const float* __restrict__ W2,    // [32, 32]
    const float* __restrict__ W3,    // [32, 1]
    const float* __restrict__ b3,    // [1]
    float* __restrict__ out,         // [B] sym | [B] ir | [B] nr | [B,81] nsc
    int B)
{
    __shared__ float Xs [TILE_B * KPAD];   // 8064 floats; reused as H2s (stride 33)
    __shared__ float W1s[(KPAD/2) * 32 * 2]; // 2688, packed K-pairs
    __shared__ float H1s[TILE_B * 34];     // 3264
    __shared__ float W2s[16 * 32 * 2];     // 1024, packed K-pairs
    __shared__ float W3s[32];
    __shared__ float irs[TILE_B];
    float* H2s = Xs;                       // alias, stride 33

    const int tid = threadIdx.x;
    const int rowBlock = blockIdx.x * TILE_B;

    // ---- Phase 0: weights into LDS, packed so B-fragment (K,K+1)@col is one b64 load
    for (int i = tid; i < (KPAD/2) * 32 * 2; i += THREADS) {   // 2688
        int k = i >> 5;          // 0..83
        int n = i & 31;
        float v = (k < STATE) ? W1[k * 32 + n] : 0.0f;
        W1s[(((k >> 1) * 32) + n) * 2 + (k & 1)] = v;
    }
    for (int i = tid; i < 1024; i += THREADS) {
        int k = i >> 5, n = i & 31;
        W2s[(((k >> 1) * 32) + n) * 2 + (k & 1)] = W2[k * 32 + n];
    }
    if (tid < 32) W3s[tid] = W3[tid];

    // ---- Phase 1: next_state_count scatter (each af element read exactly once)
    for (int i = tid; i < TILE_B * KPAD; i += THREADS) {
        int r = i / KPAD;
        int s = i - r * KPAD;
        int grow = rowBlock + r;
        float v = 0.0f;
        if (s < STATE && grow < B) {
            int rr = s / SIDE, cc = s - rr * SIDE;
            const float* af = act + (size_t)grow * (STATE * NA);
            v = af[s * NA + 0];                               // stay
            if (rr == 0)        v += af[s * NA + 1];          // up blocked
            if (rr == SIDE - 1) v += af[s * NA + 2];          // down blocked
            if (cc == 0)        v += af[s * NA + 3];          // left blocked
            if (cc == SIDE - 1) v += af[s * NA + 4];          // right blocked
            if (rr < SIDE - 1)  v += af[(s + SIDE) * NA + 1]; // inflow from below (up)
            if (rr > 0)         v += af[(s - SIDE) * NA + 2]; // inflow from above (down)
            if (cc < SIDE - 1)  v += af[(s + 1) * NA + 3];    // inflow from right (left)
            if (cc > 0)         v += af[(s - 1) * NA + 4];    // inflow from left (right)
        }
        Xs[i] = v;
    }
    __syncthreads();

    // ---- Phase 1b: immediate reward + coalesced nsc output
    if (tid < TILE_B) {
        int grow = rowBlock + tid;
        float ir = 0.0f;
        if (grow < B) {
            const float* dem = obs + (size_t)grow * (3 * STATE) + 2 * STATE;
            for (int s = 0; s < STATE; ++s)
                ir += fminf(Xs[tid * KPAD + s], dem[s]);
            out[(size_t)B + grow] = ir;
        }
        irs[tid] = ir;
    }
    for (int i = tid; i < TILE_B * STATE; i += THREADS) {
        int r = i / STATE, s = i - r * STATE;
        int grow = rowBlock + r;
        if (grow < B)
            out[(size_t)3 * B + (size_t)grow * STATE + s] = Xs[r * KPAD + s];
    }

    // ---- Phase 2: GEMM1  (Xs[96x84] @ W1[84x32]) via v_wmma_f32_16x16x4_f32
    const int wave = tid >> 5;
    const int lane = tid & 31;
    const int half = lane & 15;
    const int hi   = lane >> 4;           // lanes 16-31 hold K+2,K+3 halves
    const int rowBase = wave * 16;

    v8f c0 = {}; v8f c1 = {};
    {
        const float* arow = Xs + (rowBase + half) * KPAD + 2 * hi;
        #pragma unroll
        for (int k = 0; k < KPAD / 4; ++k) {   // 21 chunks
            v2f a  = *(const v2f*)(arow + 4 * k);
            v2f b0 = *(const v2f*)(W1s + (((2 * k + hi) * 32) + half) * 2);
            v2f b1 = *(const v2f*)(W1s + (((2 * k + hi) * 32) + half + 16) * 2);
            c0 = __builtin_amdgcn_wmma_f32_16x16x4_f32(false, a, false, b0,
                                                       (short)0, c0, false, false);
            c1 = __builtin_amdgcn_wmma_f32_16x16x4_f32(false, a, false, b1,
                                                       (short)0, c1, false, false);
        }
    }
    // ReLU + store H1 (stride 34 avoids bank conflicts)
    #pragma unroll
    for (int i = 0; i < 8; ++i) {
        int m = rowBase + i + 8 * hi;
        H1s[m * 34 + half]      = fmaxf(c0[i], 0.0f);
        H1s[m * 34 + half + 16] = fmaxf(c1[i], 0.0f);
    }
    __syncthreads();   // H1 visible; also: all Xs reads complete (safe to alias as H2s)

    // ---- Phase 3: GEMM2  (H1[96x32] @ W2[32x32])
    v8f d0 = {}; v8f d1 = {};
    {
        const float* arow = H1s + (rowBase + half) * 34 + 2 * hi;
        #pragma unroll
        for (int k = 0; k < 8; ++k) {
            v2f a  = *(const v2f*)(arow + 4 * k);
            v2f b0 = *(const v2f*)(W2s + (((2 * k + hi) * 32) + half) * 2);
            v2f b1 = *(const v2f*)(W2s + (((2 * k + hi) * 32) + half + 16) * 2);
            d0 = __builtin_amdgcn_wmma_f32_16x16x4_f32(false, a, false, b0,
                                                       (short)0, d0, false, false);
            d1 = __builtin_amdgcn_wmma_f32_16x16x4_f32(false, a, false, b1,
                                                       (short)0, d1, false, false);
        }
    }
    #pragma unroll
    for (int i = 0; i < 8; ++i) {
        int m = rowBase + i + 8 * hi;
        H2s[m * 33 + half]      = fmaxf(d0[i], 0.0f);
        H2s[m * 33 + half + 16] = fmaxf(d1[i], 0.0f);
    }
    __syncthreads();

    // ---- Phase 4: W3 head + final outputs
    if (tid < TILE_B) {
        int grow = rowBlock + tid;
        if (grow < B) {
            float acc = b3[0];
            #pragma unroll
            for (int j = 0; j < 32; ++j)
                acc += H2s[tid * 33 + j] * W3s[j];
            out[(size_t)2 * B + grow] = acc;          // next_return
            out[grow] = irs[tid] + acc;               // symbolic_val
        }
    }
}

extern "C" void kernel_launch(void* const* d_in, const int* in_sizes, int n_in,
                              void* d_out, int out_size, void* d_ws, size_t ws_size,
                              hipStream_t stream) {
    const float* obs = (const float*)d_in[0];
    const float* act = (const float*)d_in[1];
    const float* W1  = (const float*)d_in[2];
    const float* W2  = (const float*)d_in[3];
    const float* W3  = (const float*)d_in[4];
    const float* b3  = (const float*)d_in[5];
    float* out = (float*)d_out;

    int B = in_sizes[0] / (3 * STATE);
    int grid = (B + TILE_B - 1) / TILE_B;
    vpn_kernel<<<grid, THREADS, 0, stream>>>(obs, act, W1, W2, W3, b3, out, B);
}